// CoarsenBlock_37726992728632
// MI455X (gfx1250) — compile-verified
//
#include <hip/hip_runtime.h>
#include <hip/hip_bf16.h>
#include <math.h>

// Problem constants (from reference)
#define BB   8
#define NN_  2048
#define FD   256
#define KTOP 205

typedef float v2f __attribute__((ext_vector_type(2)));
typedef float v8f __attribute__((ext_vector_type(8)));
typedef int   v4i __attribute__((ext_vector_type(4)));

#if __has_builtin(__builtin_amdgcn_global_load_async_to_lds_b128)
#define HAVE_ASYNC_LDS 1
#else
#define HAVE_ASYNC_LDS 0
#endif

typedef __attribute__((address_space(1))) v4i* g_v4i_ptr;
typedef __attribute__((address_space(3))) v4i* l_v4i_ptr;

// 16-byte global->LDS copy: async DMA (ASYNCcnt) when available, else via VGPR.
__device__ __forceinline__ void copy16_g2l(const float* __restrict__ g, float* l) {
#if HAVE_ASYNC_LDS
  __builtin_amdgcn_global_load_async_to_lds_b128(
      (g_v4i_ptr)(unsigned long long)g,
      (l_v4i_ptr)(unsigned int)(unsigned long long)l,
      0, 0);
#else
  const float4 v = *reinterpret_cast<const float4*>(g);
  l[0] = v.x; l[1] = v.y; l[2] = v.z; l[3] = v.w;
#endif
}

__device__ __forceinline__ void wait_async_copies() {
#if HAVE_ASYNC_LDS
#if __has_builtin(__builtin_amdgcn_s_wait_asynccnt)
  __builtin_amdgcn_s_wait_asynccnt(0);
#else
  asm volatile("s_wait_asynccnt 0" ::: "memory");
#endif
#endif
}

// ---------------------------------------------------------------------------
// Pass 1: xw[b,n] = dot(x[b,n,:], W)        grid(N,B), block 256
// ---------------------------------------------------------------------------
__global__ __launch_bounds__(256)
void xw_kernel(const float* __restrict__ x, const float* __restrict__ W,
               float* __restrict__ xw) {
  __shared__ float sd[256];
  const int n = blockIdx.x, b = blockIdx.y, t = threadIdx.x;
  float v = x[((long)b * NN_ + n) * FD + t] * W[t];
  sd[t] = v; __syncthreads();
  for (int s = 128; s > 0; s >>= 1) { if (t < s) sd[t] += sd[t + s]; __syncthreads(); }
  if (t == 0) xw[(long)b * NN_ + n] = sd[0];
}

// ---------------------------------------------------------------------------
// Pass 2: degrees.  rs = sum_m adj[b,n,m]
//   dg = (rs+1)^-0.5 ; d = max(rs+1,1)^-0.5 ; mask = rs>0 ; dgxw = dg*xw
// ---------------------------------------------------------------------------
__global__ __launch_bounds__(256)
void degree_kernel(const float* __restrict__ adj, const float* __restrict__ xw,
                   float* __restrict__ dg, float* __restrict__ dv,
                   float* __restrict__ msk, float* __restrict__ dgxw) {
  __shared__ float sd[256];
  const int n = blockIdx.x, b = blockIdx.y, t = threadIdx.x;
  const float* row = adj + ((long)b * NN_ + n) * NN_;
  float s = 0.f;
  #pragma unroll
  for (int i = 0; i < NN_ / 256; ++i) s += row[t + i * 256];
  sd[t] = s; __syncthreads();
  for (int st = 128; st > 0; st >>= 1) { if (t < st) sd[t] += sd[t + st]; __syncthreads(); }
  if (t == 0) {
    const float rs = sd[0];
    const float ah = rs + 1.0f;
    const float dgv = rsqrtf(ah);
    const long  idx = (long)b * NN_ + n;
    dg[idx]   = dgv;
    dv[idx]   = rsqrtf(fmaxf(ah, 1.0f));
    msk[idx]  = (rs > 0.0f) ? 1.0f : 0.0f;
    dgxw[idx] = dgv * xw[idx];
  }
}

// ---------------------------------------------------------------------------
// Pass 3: alpha[b,n] = sigmoid((dg[n]*(sum_m adj[n,m]*dgxw[m] + dg[n]*xw[n]) + bias)^2)
// ---------------------------------------------------------------------------
__global__ __launch_bounds__(256)
void alpha_kernel(const float* __restrict__ adj, const float* __restrict__ dgxw,
                  const float* __restrict__ dg, const float* __restrict__ xw,
                  const float* __restrict__ bias, float* __restrict__ alpha) {
  __shared__ float sd[256];
  const int n = blockIdx.x, b = blockIdx.y, t = threadIdx.x;
  const float* row = adj + ((long)b * NN_ + n) * NN_;
  const float* dx  = dgxw + (long)b * NN_;
  float s = 0.f;
  #pragma unroll
  for (int i = 0; i < NN_ / 256; ++i) {
    const int m = t + i * 256;
    s += row[m] * dx[m];
  }
  sd[t] = s; __syncthreads();
  for (int st = 128; st > 0; st >>= 1) { if (t < st) sd[t] += sd[t + st]; __syncthreads(); }
  if (t == 0) {
    const long idx = (long)b * NN_ + n;
    const float dgn = dg[idx];
    float a = dgn * (sd[0] + dgn * xw[idx]) + bias[0];
    a = a * a;
    alpha[idx] = 1.0f / (1.0f + expf(-a));
  }
}

// ---------------------------------------------------------------------------
// Pass 4: per-batch top-K via in-LDS bitonic sort (descending, idx-stable).
//   grid(B), block 1024.  Writes cut[b] and topi (as float) into d_out slot.
// ---------------------------------------------------------------------------
__global__ __launch_bounds__(1024)
void topk_kernel(const float* __restrict__ alpha, float* __restrict__ cut,
                 float* __restrict__ topi) {
  __shared__ float v[NN_];
  __shared__ int   ix[NN_];
  const int b = blockIdx.x, t = threadIdx.x;
  const float* a = alpha + (long)b * NN_;
  v[t] = a[t];               ix[t] = t;
  v[t + 1024] = a[t + 1024]; ix[t + 1024] = t + 1024;
  __syncthreads();
  for (int k = 2; k <= NN_; k <<= 1) {
    for (int j = k >> 1; j > 0; j >>= 1) {
      const int i = (t & (j - 1)) | ((t & ~(j - 1)) << 1);
      const int p = i | j;
      const bool dirDesc = ((i & k) == 0);
      const float vi = v[i], vp = v[p];
      const int   ii = ix[i], ip = ix[p];
      // "i precedes p" in descending (value desc, index asc on ties)
      const bool iBefore = (vi > vp) || (vi == vp && ii < ip);
      const bool doSwap = dirDesc ? (!iBefore) : iBefore;
      if (doSwap) { v[i] = vp; v[p] = vi; ix[i] = ip; ix[p] = ii; }
      __syncthreads();
    }
  }
  if (t == 0) cut[b] = v[KTOP - 1];
  if (t < KTOP) topi[(long)b * KTOP + t] = (float)ix[t];
}

// ---------------------------------------------------------------------------
// Pass 5: S[b,n,m] = d[n]*(adj[n,m]+[m==n])*d[m]*mask[n]*relu(alpha[m]+1e-7-cut)
//         then row-normalized by sum |S| (clipped at 1e-12).
// ---------------------------------------------------------------------------
__global__ __launch_bounds__(256)
void s_kernel(const float* __restrict__ adj, const float* __restrict__ dv,
              const float* __restrict__ msk, const float* __restrict__ alpha,
              const float* __restrict__ cut, float* __restrict__ S) {
  __shared__ float sd[256];
  const int n = blockIdx.x, b = blockIdx.y, t = threadIdx.x;
  const float* row = adj + ((long)b * NN_ + n) * NN_;
  const float* dvb = dv + (long)b * NN_;
  const float* alb = alpha + (long)b * NN_;
  const float dn   = dvb[n];
  const float mn   = msk[(long)b * NN_ + n];
  const float cv   = cut[b];
  float vals[NN_ / 256];
  float asum = 0.f;
  #pragma unroll
  for (int i = 0; i < NN_ / 256; ++i) {
    const int m = t + i * 256;
    float ah = row[m] + ((m == n) ? 1.0f : 0.0f);
    float ca = fmaxf(alb[m] + 1e-7f - cv, 0.0f);
    float val = dn * ah * dvb[m] * mn * ca;
    vals[i] = val;
    asum += fabsf(val);
  }
  sd[t] = asum; __syncthreads();
  for (int st = 128; st > 0; st >>= 1) { if (t < st) sd[t] += sd[t + st]; __syncthreads(); }
  const float inv = 1.0f / fmaxf(sd[0], 1e-12f);
  float* srow = S + ((long)b * NN_ + n) * NN_;
  #pragma unroll
  for (int i = 0; i < NN_ / 256; ++i) srow[t + i * 256] = vals[i] * inv;
}

// ---------------------------------------------------------------------------
// WMMA f32 GEMM:  C[M,Nn] = op(A) @ B    (op = identity or transpose)
//   64x64 output tile per 256-thread block (8 waves, 4x2 of 16x32 each).
//   V_WMMA_F32_16X16X4_F32, K-blocked by 16, double-buffered LDS filled by
//   GLOBAL_LOAD_ASYNC_TO_LDS_B128 (ASYNCcnt) with one barrier per K-step:
//     wait asynccnt -> barrier -> issue next tile's async loads -> compute.
//   TA=true: A stored [Kk x M] row-major (row stride ldA), used as A^T.
//   FLOOR10K: fuse floor(c*1e4)/1e4 on store (for `coarse`).
// ---------------------------------------------------------------------------
template <bool TA, bool FLOOR10K>
__global__ __launch_bounds__(256)
void gemm_wmma_f32(const float* __restrict__ Ag, const float* __restrict__ Bg,
                   float* __restrict__ Cg, int M, int Nn, int Kk, int ldA,
                   long sA, long sB, long sC) {
  const float* A = Ag + (long)blockIdx.z * sA;
  const float* B = Bg + (long)blockIdx.z * sB;
  float*       C = Cg + (long)blockIdx.z * sC;
  const int m0 = blockIdx.y * 64;
  const int n0 = blockIdx.x * 64;

  constexpr int ASZ = TA ? 16 * 68 : 64 * 20;
  constexpr int BSZ = 16 * 68;
  __shared__ float As[2 * ASZ];
  __shared__ float Bs[2 * BSZ];

  const int t    = threadIdx.x;
  const int lane = t & 31;
  const int wid  = t >> 5;
  const int wm   = wid & 3;   // 4 wave-rows of 16
  const int wn   = wid >> 2;  // 2 wave-cols of 32
  const int half = lane >> 4; // K-half (f32 WMMA A/B layout)
  const int lr   = lane & 15;

  // One 16B chunk per thread per tile.
  const int ar = TA ? (t >> 4) : (t >> 2);
  const int ac = TA ? ((t & 15) << 2) : ((t & 3) << 2);
  const int br = t >> 4;
  const int bc = (t & 15) << 2;

  auto stage = [&](int k0, int buf) {
    float* As_ = As + buf * ASZ;
    float* Bs_ = Bs + buf * BSZ;
    if constexpr (TA) {
      copy16_g2l(A + (long)(k0 + ar) * ldA + m0 + ac, As_ + ar * 68 + ac);
    } else {
      copy16_g2l(A + (long)(m0 + ar) * ldA + k0 + ac, As_ + ar * 20 + ac);
    }
    copy16_g2l(B + (long)(k0 + br) * Nn + n0 + bc, Bs_ + br * 68 + bc);
  };

  v8f acc0 = {};
  v8f acc1 = {};

  stage(0, 0);
  int cur = 0;
  for (int k0 = 0; k0 < Kk; k0 += 16) {
    wait_async_copies();   // this wave's fills of buf `cur` are complete
    __syncthreads();       // all waves' fills complete; prev compute done
    if (k0 + 16 < Kk) {
      stage(k0 + 16, cur ^ 1);   // overlap next tile's DMA with compute
      __builtin_prefetch(B + (long)(k0 + 32 + br) * Nn + n0 + bc, 0, 1);
    }
    const float* As_ = As + cur * ASZ;
    const float* Bs_ = Bs + cur * BSZ;

    #pragma unroll
    for (int kk = 0; kk < 16; kk += 4) {
      // A fragment 16x4: lanes 0-15 hold K=kk,kk+1; lanes 16-31 hold K=kk+2,kk+3
      v2f a;
      if constexpr (TA) {
        a.x = As_[(kk + half * 2 + 0) * 68 + wm * 16 + lr];
        a.y = As_[(kk + half * 2 + 1) * 68 + wm * 16 + lr];
      } else {
        a.x = As_[(wm * 16 + lr) * 20 + kk + half * 2 + 0];
        a.y = As_[(wm * 16 + lr) * 20 + kk + half * 2 + 1];
      }
      // B fragments 4x16 for the two 16-wide n-tiles
      v2f b0, b1;
      b0.x = Bs_[(kk + half * 2 + 0) * 68 + wn * 32 + lr];
      b0.y = Bs_[(kk + half * 2 + 1) * 68 + wn * 32 + lr];
      b1.x = Bs_[(kk + half * 2 + 0) * 68 + wn * 32 + 16 + lr];
      b1.y = Bs_[(kk + half * 2 + 1) * 68 + wn * 32 + 16 + lr];
      acc0 = __builtin_amdgcn_wmma_f32_16x16x4_f32(false, a, false, b0,
                                                   (short)0, acc0, false, false);
      acc1 = __builtin_amdgcn_wmma_f32_16x16x4_f32(false, a, false, b1,
                                                   (short)0, acc1, false, false);
    }
    cur ^= 1;
  }

  // ---- store: C/D vgpr r holds M=r (lanes 0-15) and M=r+8 (lanes 16-31) ----
  #pragma unroll
  for (int r = 0; r < 8; ++r) {
    const int gm = m0 + wm * 16 + r + half * 8;
    const int gn = n0 + wn * 32 + lr;
    float v0 = acc0[r];
    float v1 = acc1[r];
    if constexpr (FLOOR10K) {
      v0 = floorf(v0 * 10000.0f) / 10000.0f;
      v1 = floorf(v1 * 10000.0f) / 10000.0f;
    }
    C[(long)gm * Nn + gn]      = v0;
    C[(long)gm * Nn + gn + 16] = v1;
  }
}

// ---------------------------------------------------------------------------
// Host launcher
// ---------------------------------------------------------------------------
extern "C" void kernel_launch(void* const* d_in, const int* in_sizes, int n_in,
                              void* d_out, int out_size, void* d_ws, size_t ws_size,
                              hipStream_t stream) {
  const float* x    = (const float*)d_in[0];   // (B,N,F)
  const float* adj  = (const float*)d_in[1];   // (B,N,N)
  const float* W    = (const float*)d_in[2];   // (F,1)
  const float* bias = (const float*)d_in[3];   // (1,)

  float* out = (float*)d_out;
  // Output tuple layout (flat, return order): x_c, coarse, S, topi
  float* x_c    = out;                                       // B*N*F
  float* coarse = out + (long)BB * NN_ * FD;                 // B*N*N
  float* S      = coarse + (long)BB * NN_ * NN_;             // B*N*N
  float* topi   = S + (long)BB * NN_ * NN_;                  // B*KTOP (as float)

  // Workspace layout: T (B*N*N) then small per-node arrays.
  float* ws   = (float*)d_ws;
  float* T    = ws;                                          // B*N*N
  float* xw   = T + (long)BB * NN_ * NN_;                    // B*N
  float* dg   = xw + (long)BB * NN_;
  float* dv   = dg + (long)BB * NN_;
  float* msk  = dv + (long)BB * NN_;
  float* dgxw = msk + (long)BB * NN_;
  float* alp  = dgxw + (long)BB * NN_;
  float* cut  = alp + (long)BB * NN_;                        // B

  const dim3 rowGrid(NN_, BB);

  // 1) xw = x @ W
  xw_kernel<<<rowGrid, 256, 0, stream>>>(x, W, xw);
  // 2) degrees / masks / dg*xw
  degree_kernel<<<rowGrid, 256, 0, stream>>>(adj, xw, dg, dv, msk, dgxw);
  // 3) alpha
  alpha_kernel<<<rowGrid, 256, 0, stream>>>(adj, dgxw, dg, xw, bias, alp);
  // 4) top-K -> cut, topi
  topk_kernel<<<dim3(BB), 1024, 0, stream>>>(alp, cut, topi);
  // 5) S (normalized assignment), written straight into its output slot
  s_kernel<<<rowGrid, 256, 0, stream>>>(adj, dv, msk, alp, cut, S);

  // 6) T = adj @ S          (NN GEMM, M=N=K=2048)
  gemm_wmma_f32<false, false><<<dim3(NN_ / 64, NN_ / 64, BB), 256, 0, stream>>>(
      adj, S, T, NN_, NN_, NN_, NN_,
      (long)NN_ * NN_, (long)NN_ * NN_, (long)NN_ * NN_);
  // 7) coarse = S^T @ T     (TN GEMM + floor(1e4) fuse)
  gemm_wmma_f32<true, true><<<dim3(NN_ / 64, NN_ / 64, BB), 256, 0, stream>>>(
      S, T, coarse, NN_, NN_, NN_, NN_,
      (long)NN_ * NN_, (long)NN_ * NN_, (long)NN_ * NN_);
  // 8) x_c = S^T @ x        (TN GEMM, Nn=256)
  gemm_wmma_f32<true, false><<<dim3(FD / 64, NN_ / 64, BB), 256, 0, stream>>>(
      S, x, x_c, NN_, FD, NN_, NN_,
      (long)NN_ * NN_, (long)NN_ * FD, (long)NN_ * FD);
}